// Generator_5669356834085
// MI455X (gfx1250) — compile-verified
//
#include <hip/hip_runtime.h>

typedef __attribute__((ext_vector_type(2))) float v2f;
typedef __attribute__((ext_vector_type(8))) float v8f;

#define IMG 128

static __device__ __forceinline__ float leakyf(float v){ return v >= 0.f ? v : 0.2f*v; }

// ------- broadcast initial block, written channels-last: xt[(b*16+p)*512 + c]
__global__ void k_bcast_t(const float* __restrict__ in, float* __restrict__ xt, int total){
  int i = blockIdx.x*blockDim.x + threadIdx.x;
  if (i >= total) return;
  int c = i % 512; int t = i / 512; int p = t % 16;
  xt[i] = in[(size_t)c*16 + p];
}

// ------- weight repack (oc,ic,3,3) -> (oc,9,ic), K contiguous for WMMA A ----
__global__ void k_wrepack(const float* __restrict__ w, float* __restrict__ wr,
                          int ic, int n){
  int i = blockIdx.x*blockDim.x + threadIdx.x;
  if (i >= n) return;                      // n = oc*ic*9
  int t = i % 9; int rem = i / 9;
  int ci = rem % ic; int o = rem / ic;
  wr[((size_t)o*9 + t)*ic + ci] = w[i];
}

// ------- style projection: out[b,c] = styles[b,blk,:] . w[:,c] + bias[c] ----
__global__ void k_style(const float* __restrict__ styles, int blk,
                        const float* __restrict__ w, const float* __restrict__ bias,
                        float* __restrict__ out, int C){
  int i = blockIdx.x*blockDim.x + threadIdx.x;
  if (i >= 8*C) return;
  int b = i / C, c = i % C;
  const float* s = styles + ((size_t)b*6 + blk)*512;
  float acc = bias[c];
  for (int k = 0; k < 512; ++k) acc = fmaf(s[k], w[(size_t)k*C + c], acc);
  out[i] = acc;
}

// ------- wsq[o,i] = sum_t w[o,i,t]^2  (n = oc*ic, original layout) ---------
__global__ void k_wsq(const float* __restrict__ w, float* __restrict__ wsq, int n){
  int i = blockIdx.x*blockDim.x + threadIdx.x;
  if (i >= n) return;
  const float* p = w + (size_t)i*9;
  float a = 0.f;
#pragma unroll
  for (int t = 0; t < 9; ++t) a = fmaf(p[t], p[t], a);
  wsq[i] = a;
}

// ------- demod d[b,o] = rsqrt(sum_i (s+1)^2 * wsq[o,i] + eps) --------------
__global__ void k_demod(const float* __restrict__ s, const float* __restrict__ wsq,
                        float* __restrict__ d, int ic, int oc){
  int i = blockIdx.x*blockDim.x + threadIdx.x;
  if (i >= 8*oc) return;
  int b = i / oc, o = i % oc;
  const float* sb = s + (size_t)b*ic;
  const float* wo = wsq + (size_t)o*ic;
  float a = 0.f;
  for (int k = 0; k < ic; ++k){ float m = sb[k] + 1.f; a = fmaf(m*m, wo[k], a); }
  d[i] = rsqrtf(a + 1e-8f);
}

// ------- modulate + transpose: xt[(b*HW+p)*C + c] = x[(b,c,p)] * (s+1) -----
__global__ void k_modscale_t(const float* __restrict__ x, const float* __restrict__ s,
                             float* __restrict__ xt, int C, int HW, int total){
  int i = blockIdx.x*blockDim.x + threadIdx.x;
  if (i >= total) return;
  int c = i % C; int t = i / C;
  int p = t % HW; int b = t / HW;
  xt[i] = x[((size_t)b*C + c)*HW + p] * (s[(size_t)b*C + c] + 1.f);
}

// ------- jax bilinear 2x (half-pixel, edge-renorm) + modulate + transpose --
// output: xt[((b*(2H*2W)) + r2*2W + c2) * C + ch]
__global__ void k_up2x_t(const float* __restrict__ in, float* __restrict__ xt,
                         const float* __restrict__ s, int C, int H, int W, int total){
  int i = blockIdx.x*blockDim.x + threadIdx.x;
  if (i >= total) return;
  int ch = i % C; int t = i / C;
  int Wo = 2*W, Ho = 2*H;
  int c2 = t % Wo; t /= Wo;
  int r2 = t % Ho; int b = t / Ho;
  const float* p = in + ((size_t)b*C + ch) * H * W;
  int ra = r2 >> 1; int rb = (r2 & 1) ? ra + 1 : ra - 1;
  bool rv = (rb >= 0) && (rb < H);
  int ca = c2 >> 1; int cb = (c2 & 1) ? ca + 1 : ca - 1;
  bool cv = (cb >= 0) && (cb < W);
  auto rowval = [&](int rr) -> float {
    float m = p[(size_t)rr*W + ca];
    return cv ? fmaf(0.25f, p[(size_t)rr*W + cb], 0.75f*m) : m;
  };
  float v = rv ? fmaf(0.25f, rowval(rb), 0.75f*rowval(ra)) : rowval(ra);
  xt[i] = v * (s[(size_t)b*C + ch] + 1.f);
}

// ------- plain bilinear 2x for RGB (NCHW in/out) ---------------------------
__global__ void k_up2x(const float* __restrict__ in, float* __restrict__ out,
                       int C, int H, int W, int total){
  int i = blockIdx.x*blockDim.x + threadIdx.x;
  if (i >= total) return;
  int Wo = 2*W, Ho = 2*H;
  int c2 = i % Wo; int t = i / Wo;
  int r2 = t % Ho; t /= Ho;
  int ch = t % C;  int b = t / C;
  const float* p = in + ((size_t)b*C + ch) * H * W;
  int ra = r2 >> 1; int rb = (r2 & 1) ? ra + 1 : ra - 1;
  bool rv = (rb >= 0) && (rb < H);
  int ca = c2 >> 1; int cb = (c2 & 1) ? ca + 1 : ca - 1;
  bool cv = (cb >= 0) && (cb < W);
  auto rowval = [&](int rr) -> float {
    float m = p[(size_t)rr*W + ca];
    return cv ? fmaf(0.25f, p[(size_t)rr*W + cb], 0.75f*m) : m;
  };
  out[i] = rv ? fmaf(0.25f, rowval(rb), 0.75f*rowval(ra)) : rowval(ra);
}

// ------- depthwise 3x3 blur, kernel (1,2,1)x(1,2,1)/16, zero pad -----------
__global__ void k_blur(const float* __restrict__ in, float* __restrict__ out,
                       int C, int H, int W, int total){
  int i = blockIdx.x*blockDim.x + threadIdx.x;
  if (i >= total) return;
  int c2 = i % W; int t = i / W;
  int r2 = t % H; t /= H;
  int ch = t % C; int b = t / C;
  const float* p = in + ((size_t)b*C + ch) * H * W;
  const float kw[3] = {1.f, 2.f, 1.f};
  float acc = 0.f;
#pragma unroll
  for (int dh = -1; dh <= 1; ++dh){
#pragma unroll
    for (int dw = -1; dw <= 1; ++dw){
      int rr = r2 + dh, cc = c2 + dw;
      if (rr >= 0 && rr < H && cc >= 0 && cc < W)
        acc = fmaf(kw[dh+1]*kw[dw+1], p[(size_t)rr*W + cc], acc);
    }
  }
  out[i] = acc * (1.f/16.f);
}

// ------- to-RGB 1x1 conv (3 out ch, modulated, no demod) + skip ------------
__global__ void k_rgb(const float* __restrict__ x, const float* __restrict__ s,
                      const float* __restrict__ w, const float* __restrict__ prev,
                      float* __restrict__ out, int C, int HW, int total){
  int i = blockIdx.x*blockDim.x + threadIdx.x;
  if (i >= total) return;
  int p = i % HW; int t = i / HW;
  int k = t % 3;  int b = t / 3;
  float acc = prev ? prev[i] : 0.f;
  const float* xb = x + (size_t)b*C*HW + p;
  const float* wk = w + (size_t)k*C;
  const float* sb = s + (size_t)b*C;
  for (int c = 0; c < C; ++c)
    acc = fmaf(wk[c]*(sb[c] + 1.f), xb[(size_t)c*HW], acc);
  out[i] = acc;
}

// ------- implicit-GEMM 3x3 conv via V_WMMA_F32_16X16X4_F32 -----------------
// Both operands are K-innermost: A from wr (oc,9,ic), B from xt (b,HW,ic),
// so each lane fetches its K-pair with one aligned b64 load per operand.
// 4 waves per workgroup, each owning a 16(oc) x 16(pixel) f32 tile.
// A: lane L -> M=L%16, K=2*(L/16)+{0,1}; B: lane L -> N=L%16, same K pair;
// D: vgpr r, lane L -> (M = r+8*(L/16), N = L%16).
// mode 0: y = acc + bias      (initial conv)
// mode 1: y = leaky(demod[b,o]*acc + noise[b,col,row]*nw[o] + nb[o])
__global__ __launch_bounds__(128)
void k_conv3x3(const float* __restrict__ xt,    // (8, HW, ic) channels-last
               const float* __restrict__ wr,    // (oc, 9, ic) repacked
               float* __restrict__ y,           // (8, oc, H, W)
               const float* __restrict__ dvec,  // (8, oc)
               const float* __restrict__ bias,  // (oc)
               const float* __restrict__ noise, // (8,128,128,1)
               const float* __restrict__ nw, const float* __restrict__ nb,
               int ic, int oc, int H, int W, int mode){
  const int lane = threadIdx.x & 31;
  const int wv   = threadIdx.x >> 5;
  const int half = lane >> 4;
  const int l16  = lane & 15;
  const int b     = blockIdx.z;
  const int otile = blockIdx.y;
  const int HW = H * W;
  const int ptile = blockIdx.x * 4 + wv;
  if (ptile * 16 >= HW) return;              // wave-uniform: EXEC stays full
  const int pos = ptile * 16 + l16;
  const int row = pos / W, col = pos % W;
  const int oA = otile * 16 + l16;
  const float* xb = xt + (size_t)b * HW * ic;
  const float* wb = wr + (size_t)oA * 9 * ic;
  const int k0 = 2 * half;                   // this lane's K-pair base

  v8f acc = {};
  for (int tap = 0; tap < 9; ++tap){
    const int dh = tap / 3 - 1, dw = tap % 3 - 1;
    const int r2 = row + dh, c2 = col + dw;
    const bool valid = (r2 >= 0) && (r2 < H) && (c2 >= 0) && (c2 < W);
    const float msk = valid ? 1.f : 0.f;     // zero padding via mask
    const int pos2 = valid ? (r2 * W + c2) : 0;
    const float* xp = xb + (size_t)pos2 * ic + k0;
    const float* wp = wb + (size_t)tap * ic + k0;
    for (int kk = 0; kk < ic; kk += 4){
      v2f a  = *(const v2f*)(wp + kk);       // one b64 load (A K-pair)
      v2f bv = *(const v2f*)(xp + kk);       // one b64 load (B K-pair)
      bv.x *= msk; bv.y *= msk;
      acc = __builtin_amdgcn_wmma_f32_16x16x4_f32(
              false, a, false, bv, (short)0, acc, false, false);
    }
  }

  union { v8f v; float f[8]; } u; u.v = acc;
  const int ob = otile * 16 + 8 * half;
  if (mode == 1){
    const float nval = noise[(size_t)b * IMG * IMG + (size_t)col * IMG + row];
#pragma unroll
    for (int r = 0; r < 8; ++r){
      const int o = ob + r;
      float v = u.f[r] * dvec[(size_t)b * oc + o] + nval * nw[o] + nb[o];
      y[((size_t)b * oc + o) * HW + pos] = leakyf(v);
    }
  } else {
#pragma unroll
    for (int r = 0; r < 8; ++r){
      const int o = ob + r;
      y[((size_t)b * oc + o) * HW + pos] = u.f[r] + bias[o];
    }
  }
}

// ============================ host orchestration ===========================
extern "C" void kernel_launch(void* const* d_in, const int* in_sizes, int n_in,
                              void* d_out, int out_size, void* d_ws, size_t ws_size,
                              hipStream_t stream){
  (void)in_sizes; (void)n_in; (void)out_size; (void)ws_size;
  const float* styles  = (const float*)d_in[0];   // (8, 6, 512)
  const float* noise   = (const float*)d_in[1];   // (8, 128, 128, 1)
  const float* initblk = (const float*)d_in[2];   // (1, 512, 4, 4)
  const float* initw   = (const float*)d_in[3];   // (512, 512, 3, 3)
  const float* initb   = (const float*)d_in[4];   // (512)

  float* ws    = (float*)d_ws;
  float* XT    = ws;                  // channels-last conv input: 8*128*128*64
  float* XA    = XT + 8388608u;       // conv output ping: 8*32*128*128
  float* XB    = XA + 4194304u;       // conv output pong
  float* WR    = XB + 4194304u;       // repacked weights: 512*512*9
  float* RGB   = WR + 2359296u;       // current-res rgb: 8*3*128*128
  float* RGBUP = RGB + 393216u;       // upsampled rgb (pre-blur)
  float* RGBP  = RGBUP + 393216u;     // blurred prev rgb for next block
  float* S1    = RGBP + 393216u;      // (8, 512)
  float* S2    = S1 + 4096u;
  float* SR    = S2 + 4096u;
  float* DV    = SR + 4096u;          // demod (8, 512)
  float* WSQ   = DV + 4096u;          // (512, 512)

  auto cdiv = [](int a, int b){ return (a + b - 1) / b; };

  // ---- initial: broadcast (transposed) + plain conv3x3 (+bias), 4x4 ----
  {
    int total = 8 * 512 * 16;
    k_bcast_t<<<cdiv(total,256),256,0,stream>>>(initblk, XT, total);
    k_wrepack<<<cdiv(512*512*9,256),256,0,stream>>>(initw, WR, 512, 512*512*9);
    dim3 g(1, 512/16, 8);
    k_conv3x3<<<g,128,0,stream>>>(XT, WR, XA, nullptr, initb,
                                  nullptr, nullptr, nullptr, 512, 512, 4, 4, 0);
  }

  static const int ICB[6]   = {512,512,512,256,128,64};
  static const int OCB[6]   = {512,512,256,128,64,32};
  static const int UP[6]    = {0,1,1,1,1,1};
  static const int UPRGB[6] = {1,1,1,1,1,0};

  float* x = XA;   // current activation (NCHW); always XA after each block
  int H = 4;
  for (int i = 0; i < 6; ++i){
    const int ic = ICB[i], oc = OCB[i];
    const int base = 5 + 13*i;
    const float* s1w = (const float*)d_in[base+0];
    const float* s1b = (const float*)d_in[base+1];
    const float* n1w = (const float*)d_in[base+2];
    const float* n1b = (const float*)d_in[base+3];
    const float* c1w = (const float*)d_in[base+4];
    const float* s2w = (const float*)d_in[base+5];
    const float* s2b = (const float*)d_in[base+6];
    const float* n2w = (const float*)d_in[base+7];
    const float* n2b = (const float*)d_in[base+8];
    const float* c2w = (const float*)d_in[base+9];
    const float* srw = (const float*)d_in[base+10];
    const float* srb = (const float*)d_in[base+11];
    const float* rgw = (const float*)d_in[base+12];

    k_style<<<cdiv(8*ic,256),256,0,stream>>>(styles, i, s1w, s1b, S1, ic);
    k_style<<<cdiv(8*oc,256),256,0,stream>>>(styles, i, s2w, s2b, S2, oc);
    k_style<<<cdiv(8*oc,256),256,0,stream>>>(styles, i, srw, srb, SR, oc);

    if (UP[i]) {
      int total = 8 * ic * (2*H) * (2*H);
      k_up2x_t<<<cdiv(total,256),256,0,stream>>>(x, XT, S1, ic, H, H, total);
      H *= 2;
    } else {
      int total = 8 * ic * H * H;
      k_modscale_t<<<cdiv(total,256),256,0,stream>>>(x, S1, XT, ic, H*H, total);
    }
    const int HW = H * H;

    // conv1: ic -> oc
    k_wrepack<<<cdiv(oc*ic*9,256),256,0,stream>>>(c1w, WR, ic, oc*ic*9);
    k_wsq<<<cdiv(oc*ic,256),256,0,stream>>>(c1w, WSQ, oc*ic);
    k_demod<<<cdiv(8*oc,256),256,0,stream>>>(S1, WSQ, DV, ic, oc);
    dim3 g(cdiv(HW/16,4), oc/16, 8);
    k_conv3x3<<<g,128,0,stream>>>(XT, WR, XB, DV, nullptr, noise, n1w, n1b,
                                  ic, oc, H, H, 1);

    // conv2: oc -> oc
    {
      int total = 8 * oc * HW;
      k_modscale_t<<<cdiv(total,256),256,0,stream>>>(XB, S2, XT, oc, HW, total);
    }
    k_wrepack<<<cdiv(oc*oc*9,256),256,0,stream>>>(c2w, WR, oc, oc*oc*9);
    k_wsq<<<cdiv(oc*oc,256),256,0,stream>>>(c2w, WSQ, oc*oc);
    k_demod<<<cdiv(8*oc,256),256,0,stream>>>(S2, WSQ, DV, oc, oc);
    k_conv3x3<<<g,128,0,stream>>>(XT, WR, x, DV, nullptr, noise, n2w, n2b,
                                  oc, oc, H, H, 1);

    // to-RGB + skip; final block writes d_out directly
    float* rgbdst = (i == 5) ? (float*)d_out : RGB;
    int totalr = 8 * 3 * HW;
    k_rgb<<<cdiv(totalr,256),256,0,stream>>>(x, SR, rgw,
                                             (i == 0) ? nullptr : RGBP,
                                             rgbdst, oc, HW, totalr);
    if (UPRGB[i]) {
      int totup = 8 * 3 * (2*H) * (2*H);
      k_up2x<<<cdiv(totup,256),256,0,stream>>>(rgbdst, RGBUP, 3, H, H, totup);
      k_blur<<<cdiv(totup,256),256,0,stream>>>(RGBUP, RGBP, 3, 2*H, 2*H, totup);
    }
  }
}